// OTAMMatching_58402965291157
// MI455X (gfx1250) — compile-verified
//
#include <hip/hip_runtime.h>

// ---------------------------------------------------------------------------
// OTAM soft-DTW matching for MI455X (gfx1250, wave32).
//
//  - 1 thread = 1 (query,support) pair; forward DP (row-streamed) and
//    transposed DP (column-streamed) in a single pass -> 256 MB read once
//    (~11us HBM floor at 23.3 TB/s).
//  - gfx1250 async tensor path: global_load_async_to_lds_b128 staged through
//    4 LDS buffers, 3 rows in flight per wave, s_wait_asynccnt pipelining.
//  - TRANS unit is the tightest resource (~0.35G exp/log lanes): softmin
//    uses base-2 hardware v_exp_f32 / v_log_f32 with folded constants,
//    min3/med3/max3 to drop the always-1 exp term at boundary cells.
// ---------------------------------------------------------------------------

#define TQ 16
#define TS 16
#define W  (TS + 2)              // padded DP width = 18
#define BLOCK 256
#define ROW_STRIDE_B 80          // 64B row + 16B pad (<=2-way bank conflict on b128)
#define BUF_B (BLOCK * ROW_STRIDE_B)   // 20480 B per buffer
#define NBUF 4                   // 4 buffers = 80 KB LDS -> 4 blocks/WGP

// softmin_l(...) = -l*ln(sum e^{-x/l}), l = 0.1
// base-2 folded: K = (1/l)*log2(e), L = l*ln(2)
#define SM_K 14.426950408889634f
#define SM_L 0.06931471805599453f

__device__ __forceinline__ float softmin2(float a, float b) {
    float mn = fminf(a, b);
    // |a-b| folds into a VALU source modifier; one exp term is exactly 1.
    float s  = 1.0f + __builtin_amdgcn_exp2f(-SM_K * fabsf(a - b));
    return fmaf(-SM_L, __builtin_amdgcn_logf(s), mn);
}

__device__ __forceinline__ float softmin3(float a, float b, float c) {
    float mn = fminf(fminf(a, b), c);            // v_min3_num_f32
    float md = __builtin_amdgcn_fmed3f(a, b, c); // v_med3_num_f32
    float mx = fmaxf(fmaxf(a, b), c);            // v_max3_num_f32
    // min term is exp2(0)=1 -> only 2 exps + 1 log (3 TRANS, was 4).
    float s  = 1.0f + __builtin_amdgcn_exp2f((mn - md) * SM_K)
                    + __builtin_amdgcn_exp2f((mn - mx) * SM_K);
    return fmaf(-SM_L, __builtin_amdgcn_logf(s), mn);
}

// Stage one 64B tile row into LDS. INST_OFFSET applies to BOTH the LDS and
// global address on gfx1250 async loads, so 4 instructions share operands.
__device__ __forceinline__ void async_row_load(unsigned ldsOff, const float* g) {
    asm volatile(
        "global_load_async_to_lds_b128 %0, %1, off\n\t"
        "global_load_async_to_lds_b128 %0, %1, off offset:16\n\t"
        "global_load_async_to_lds_b128 %0, %1, off offset:32\n\t"
        "global_load_async_to_lds_b128 %0, %1, off offset:48"
        :: "v"(ldsOff), "v"(g)
        : "memory");
}

__device__ __forceinline__ void read_row(const unsigned char* lp, float* d) {
    const float4* q = (const float4*)lp;
    float4 a = q[0], b = q[1], c = q[2], e = q[3];
    d[0]  = 1.0f - a.x; d[1]  = 1.0f - a.y; d[2]  = 1.0f - a.z; d[3]  = 1.0f - a.w;
    d[4]  = 1.0f - b.x; d[5]  = 1.0f - b.y; d[6]  = 1.0f - b.z; d[7]  = 1.0f - b.w;
    d[8]  = 1.0f - c.x; d[9]  = 1.0f - c.y; d[10] = 1.0f - c.z; d[11] = 1.0f - c.w;
    d[12] = 1.0f - e.x; d[13] = 1.0f - e.y; d[14] = 1.0f - e.z; d[15] = 1.0f - e.w;
}

__global__ __launch_bounds__(BLOCK) void otam_kernel(const float* __restrict__ sim,
                                                     float* __restrict__ out,
                                                     int npairs) {
    extern __shared__ unsigned char smem[];
    const int tid = threadIdx.x;
    const int p   = blockIdx.x * BLOCK + tid;
    if (p >= npairs) return;

    const float* gbase = sim + (size_t)p * (TQ * TS);
    // low 32 bits of a flat LDS pointer == byte offset from the workgroup's
    // LDS base; exactly what the async instruction's VDST operand wants.
    const unsigned ldsBase = (unsigned)(size_t)smem + (unsigned)(tid * ROW_STRIDE_B);

    float prevF[W];   // forward DP: previous row (prevF[0] == 0 always)
    float colT[TS];   // transposed DP: previous column
    float drow[TS];

    // Prologue: rows 0,1,2 in flight (3-deep prefetch, buffers 0..2).
    async_row_load(ldsBase + 0u * BUF_B, gbase);
    async_row_load(ldsBase + 1u * BUF_B, gbase + TS);
    async_row_load(ldsBase + 2u * BUF_B, gbase + 2 * TS);

    // -------- consume streamed row 0 --------
    asm volatile("s_wait_asynccnt 8" ::: "memory");   // rows 1,2 still in flight
    read_row(smem + (size_t)tid * ROW_STRIDE_B, drow);

    // Forward DP row 0: cumulative sum over padded width.
    {
        float acc = 0.0f;
        prevF[0] = 0.0f;
        #pragma unroll
        for (int j = 0; j < TS; ++j) { acc += drow[j]; prevF[j + 1] = acc; }
        prevF[W - 1] = acc;                       // right pad column (d = 0)
    }
    // Transposed DP column m = 1 (boundary column): dT[l,1] = d[0,l] = drow[l].
    {
        float above = drow[0];                    // cumT[0,1] = 0 + dT[0,1]
        colT[0] = above;
        #pragma unroll
        for (int l = 1; l < TS; ++l) {
            float v = drow[l] + softmin3(0.0f, 0.0f, above);
            colT[l] = v;
            above = v;
        }
    }

    // -------- streamed rows 1..15 --------
    #pragma unroll 1
    for (int r = 1; r < TQ; ++r) {
        const int nxt = r + 2;                    // keep 3 rows in flight
        if (nxt < TQ)
            async_row_load(ldsBase + (unsigned)((nxt & (NBUF - 1)) * BUF_B),
                           gbase + nxt * TS);
        // Async loads complete in order: wait until only rows > r remain.
        if (r <= TQ - 3)       asm volatile("s_wait_asynccnt 8" ::: "memory");
        else if (r == TQ - 2)  asm volatile("s_wait_asynccnt 4" ::: "memory");
        else                   asm volatile("s_wait_asynccnt 0" ::: "memory");

        read_row(smem + (size_t)((r & (NBUF - 1)) * BUF_B)
                      + (size_t)tid * ROW_STRIDE_B, drow);

        // Forward DP row step: cum[m] = d + softmin(up-left, left [, up at m=1,17]).
        {
            float carry   = 0.0f;   // cum[r, m-1], cum[r,0] = 0
            float leftOld = 0.0f;   // cum[r-1, m-1] = prevF[0]
            #pragma unroll
            for (int m = 1; m <= W - 1; ++m) {
                float curOld = prevF[m];                       // cum[r-1, m]
                float dval   = (m <= TS) ? drow[m - 1] : 0.0f; // right pad: d = 0
                float v = (m == 1 || m == W - 1)
                              ? dval + softmin3(leftOld, carry, curOld)
                              : dval + softmin2(leftOld, carry);
                prevF[m] = v;
                leftOld  = curOld;
                carry    = v;
            }
        }
        // Transposed DP column m = r+1 (interior, 2..16): dT[l,m] = d[r,l].
        {
            float diagOld = colT[0];            // cumT[l-1, m-1]
            colT[0] = colT[0] + drow[0];        // top row is a cumsum
            #pragma unroll
            for (int l = 1; l < TS; ++l) {
                float curOld = colT[l];          // cumT[l, m-1]
                float v = drow[l] + softmin2(diagOld, curOld);
                colT[l] = v;
                diagOld = curOld;
            }
        }
    }

    // Transposed DP final pad column m = 17 (boundary, d = 0).
    {
        float diagOld = colT[0];
        float above   = colT[0];                // cumT[0,17] = cumT[0,16] + 0
        #pragma unroll
        for (int l = 1; l < TS; ++l) {
            float curOld = colT[l];
            float v = softmin3(diagOld, curOld, above);
            colT[l] = v;
            diagOld = curOld;
            above   = v;
        }
    }

    out[p] = -0.5f * (prevF[W - 1] + colT[TS - 1]);
}

extern "C" void kernel_launch(void* const* d_in, const int* in_sizes, int n_in,
                              void* d_out, int out_size, void* d_ws, size_t ws_size,
                              hipStream_t stream) {
    (void)in_sizes; (void)n_in; (void)d_ws; (void)ws_size;
    const float* sim = (const float*)d_in[0];
    float* out = (float*)d_out;
    const int npairs = out_size;                       // 2048 * 128
    const int grid   = (npairs + BLOCK - 1) / BLOCK;
    otam_kernel<<<grid, BLOCK, NBUF * BUF_B, stream>>>(sim, out, npairs);
}